// SphericalHarmonics_3659312136706
// MI455X (gfx1250) — compile-verified
//
#include <hip/hip_runtime.h>
#include <stdint.h>

#define NMAX 20
#define NCH  (2 * (NMAX + 1) * (NMAX + 1))   /* 882 channels */
#define WMAXC 126                            /* widest chunk (rows 9..11) */
#define SLAB_DW (32 * WMAXC)                 /* dwords per wave slab buffer */
#define WAVES_PER_BLOCK 2
#define BLOCK (WAVES_PER_BLOCK * 32)
#define NCHUNK 13

typedef unsigned int v4u __attribute__((ext_vector_type(4)));
typedef int          v8i __attribute__((ext_vector_type(8)));
typedef int          v4i __attribute__((ext_vector_type(4)));

// Chunks of consecutive Legendre rows; odd row-count per chunk keeps chunk width
// W = 2 mod 4 -> gcd(W,64)=2 -> 32 lanes hit distinct LDS banks (conflict-free).
__device__ constexpr int chunk_first[NCHUNK] = {0, 3, 6, 9, 12, 13, 14, 15, 16, 17, 18, 19, 20};
__device__ constexpr int chunk_last [NCHUNK] = {2, 5, 8, 11, 12, 13, 14, 15, 16, 17, 18, 19, 20};

// ---- compile-time sqrt so all recurrence coefficients fold to f32 literals ----
constexpr double csqrt(double v) {
    double r = v > 1.0 ? v : 1.0;
    for (int i = 0; i < 200; ++i) r = 0.5 * (r + v / r);
    return r;
}

struct Tab {
    float a[NMAX + 1][NMAX + 1];  // a(n,m): general recurrence, valid m <= n-2
    float b[NMAX + 1][NMAX + 1];  // b(n,m)
    float e[NMAX + 1];            // sqrt(2n+1): m = n-1 step
    float d[NMAX + 1];            // -sqrt((2n+1)/(2n)): diagonal step
};
constexpr Tab mk_tab() {
    Tab t{};
    for (int n = 0; n <= NMAX; ++n) {
        t.e[n] = (float)csqrt(2.0 * n + 1.0);
        t.d[n] = n > 0 ? (float)(0.0 - csqrt((2.0 * n + 1.0) / (2.0 * n))) : 0.0f;
        for (int m = 0; m + 2 <= n; ++m) {
            double N = n, M = m;
            t.a[n][m] = (float)csqrt((4.0 * N * N - 1.0) / (N * N - M * M));
            t.b[n][m] = (float)csqrt((2.0 * N + 1.0) * (N - 1.0 - M) * (N - 1.0 + M) /
                                     ((2.0 * N - 3.0) * (N * N - M * M)));
        }
    }
    return t;
}
static constexpr Tab T = mk_tab();

// ---- TDM issue: one 2D tensor store per wave per channel-chunk ----
__device__ __forceinline__ void tdm_store_chunk(uint32_t lds_addr, uint64_t gaddr,
                                                int W /* floats per point chunk */) {
    // D# group0: count=1, lds_addr, 57-bit global addr, type=2
    v4u g0;
    g0.x = 1u;
    g0.y = lds_addr;
    g0.z = (uint32_t)gaddr;
    g0.w = ((uint32_t)(gaddr >> 32) & 0x01FFFFFFu) | (2u << 30);
    // D# group1: data_size=4B, huge tensor dims (never OOB), tile W x 32, stride 882
    v8i g1;
    g1.s0 = 0x20000;                      // workgroup_mask=0, data_size=2 (4 bytes)
    g1.s1 = 0;                            // atomic_barrier_addr=0, tensor_dim0 lo16=0 (dim0 = 1<<30)
    g1.s2 = 0x4000;                       // tensor_dim0 hi | tensor_dim1 lo (both 1<<30)
    g1.s3 = 0x4000 | (W << 16);           // tensor_dim1 hi | tile_dim0 = W
    g1.s4 = 32;                           // tile_dim1 = 32 points, tile_dim2 = 0 (2D)
    g1.s5 = NCH;                          // tensor_dim0_stride = 882 elements
    g1.s6 = 0;
    g1.s7 = 0;
    // Benign non-restrictive higher dims (tensor_dim2..4 = 1, tiles unused)
    v4i g2; g2.x = 1; g2.y = 1; g2.z = 0; g2.w = 0;
    v4i g3; g3.x = 0; g3.y = 0x00010000; g3.z = 0; g3.w = 0;

#if __has_builtin(__builtin_amdgcn_tensor_store_from_lds)
#if __clang_major__ >= 23
    v8i gz8 = {0, 0, 0, 0, 0, 0, 0, 0};
    __builtin_amdgcn_tensor_store_from_lds(g0, g1, g2, g3, gz8, 0);
#else
    __builtin_amdgcn_tensor_store_from_lds(g0, g1, g2, g3, 0);
#endif
#else
    asm volatile("tensor_store_from_lds %0, %1, %2, %3"
                 :: "s"(g0), "s"(g1), "s"(g2), "s"(g3) : "memory");
#endif
}

__global__ __launch_bounds__(BLOCK) void sh_tdm_kernel(const float* __restrict__ theta_g,
                                                       const float* __restrict__ phi_g,
                                                       float* __restrict__ out,
                                                       int n_points) {
    __shared__ float lds[WAVES_PER_BLOCK * 2 * SLAB_DW];   // 64512 bytes

    const int tid  = threadIdx.x;
    const int lane = tid & 31;
    const int wave = __builtin_amdgcn_readfirstlane(tid >> 5);
    const int wave_base = blockIdx.x * BLOCK + wave * 32;
    if (wave_base >= n_points) return;
    const int p = wave_base + lane;

    const float th = theta_g[p];
    const float ph = phi_g[p];
    const float x  = cosf(ph);
    const float s  = sqrtf(fmaxf(1.0f - x * x, 0.0f));
    float s1, c1;
    sincosf(th, &s1, &c1);

    // cos(m*theta), sin(m*theta) by angle addition (fully in registers)
    float cm[NMAX + 1], sm[NMAX + 1];
    cm[0] = 1.0f; sm[0] = 0.0f;
#pragma unroll
    for (int m = 1; m <= NMAX; ++m) {
        cm[m] = cm[m - 1] * c1 - sm[m - 1] * s1;
        sm[m] = sm[m - 1] * c1 + cm[m - 1] * s1;
    }

    // Rolling Legendre state: A[m] = Pbar_m^{n}, Bv[m] = Pbar_m^{n-1}
    float A[NMAX + 1], Bv[NMAX + 1];
#pragma unroll
    for (int m = 0; m <= NMAX; ++m) { A[m] = 0.0f; Bv[m] = 0.0f; }
    float diag = 0.28209479177387814f;   // 1/sqrt(4*pi)
    A[0] = diag;

    const uint64_t out_base = (uint64_t)(uintptr_t)(out + (size_t)wave_base * NCH);

#pragma unroll
    for (int ck = 0; ck < NCHUNK; ++ck) {
        const int nf = chunk_first[ck];
        const int nl = chunk_last[ck];
        const int c0 = 2 * nf * nf;                       // first channel of chunk
        const int W  = 2 * (nl + 1) * (nl + 1) - c0;      // chunk width in floats

        // Reusing buffer (ck&1): the TDM store issued 2 chunks ago must have drained.
        // TDM ops complete in order per wave -> tensorcnt<=1 means chunk ck-2 is done.
        if (ck >= 2) asm volatile("s_wait_tensorcnt 1" ::: "memory");

        float* slab = &lds[(wave * 2 + (ck & 1)) * SLAB_DW];
        const int lbase = lane * W;                       // packed [point][W], stride W dwords

#pragma unroll
        for (int n = nf; n <= nl; ++n) {
            if (n >= 1) {
#pragma unroll
                for (int m = 0; m <= n - 2; ++m) {
                    float Pn = T.a[n][m] * x * A[m] - T.b[n][m] * Bv[m];
                    Bv[m] = A[m];
                    A[m]  = Pn;
                }
                {   // m = n-1: Pbar_{n-1}^{n} = sqrt(2n+1) * x * Pbar_{n-1}^{n-1}
                    float Pn = T.e[n] * x * A[n - 1];
                    Bv[n - 1] = A[n - 1];
                    A[n - 1]  = Pn;
                }
                diag = T.d[n] * s * diag;    // diagonal Pbar_n^n
                A[n] = diag;
            }

            const int roff = 2 * n * n - c0;              // row offset inside chunk
#pragma unroll
            for (int mm = -n; mm <= n; ++mm) {
                const int am = mm < 0 ? -mm : mm;
                float Pv = A[am];
                float re = Pv * cm[am];
                float im = Pv * sm[am];
                if (mm < 0) {                // Y_n^{-m} = (-1)^m conj(Y_n^m)
                    if (am & 1) re = -re;    // sgn=-1: re flips, im = -sgn*im = +im
                    else        im = -im;    // sgn=+1: im flips
                }
                slab[lbase + roff + 2 * (n + mm)]     = re;
                slab[lbase + roff + 2 * (n + mm) + 1] = im;
            }
        }

        // Make this wave's LDS writes visible before the TDM reads the slab.
        asm volatile("s_wait_dscnt 0" ::: "memory");

        uint32_t lds_addr = __builtin_amdgcn_readfirstlane(
            (uint32_t)(uintptr_t)slab);  // low 32 bits of flat LDS ptr == DS offset
        uint64_t gaddr = out_base + (uint64_t)c0 * 4u;

        tdm_store_chunk(lds_addr, gaddr, W);
    }
    // S_ENDPGM implicitly waits for all counters (incl. TENSORcnt) -> stores complete.
}

extern "C" void kernel_launch(void* const* d_in, const int* in_sizes, int n_in,
                              void* d_out, int out_size, void* d_ws, size_t ws_size,
                              hipStream_t stream) {
    const float* theta = (const float*)d_in[0];
    const float* phi   = (const float*)d_in[1];
    float*       out   = (float*)d_out;
    const int n_points = in_sizes[0];              // 16 * 4096 = 65536
    const int blocks   = (n_points + BLOCK - 1) / BLOCK;
    sh_tdm_kernel<<<blocks, BLOCK, 0, stream>>>(theta, phi, out, n_points);
}